// SPDUnTangentSpace_21328807592138
// MI455X (gfx1250) — compile-verified
//
#include <hip/hip_runtime.h>

typedef __attribute__((ext_vector_type(2))) float v2f;
typedef __attribute__((ext_vector_type(8))) float v8f;

#define MATN   128
#define LDSS   132                // padded LDS row stride (banks: 132 % 64 = 4)
#define MATEL  (MATN * LDSS)      // floats per LDS matrix buffer
#define VECLEN 8256               // 128*129/2

// ---------------------------------------------------------------------------
// Workgroup GEMM: D = A * B, all three 128x128 matrices resident in LDS.
// 8 waves; wave w computes the 16-row band tm = w (8 tiles of 16x16).
// Register blocking: the wave's whole 16x128 A band is hoisted into 64 VGPRs
// (32 x v2f fragments) once, then each of the 8 N-tiles only streams B
// fragments from LDS -> ~1 ds op per WMMA instead of ~2.
// Fragment layouts per CDNA5 ISA 7.12.2:
//   A (16x4 f32):  lane L holds row M = L%16; K = vgpr + 2*(L>=16)
//   B (4x16 f32):  lane L holds col N = L%16; K = vgpr + 2*(L>=16)
//   C/D (16x16):   lane L, vgpr r -> row = r + 8*(L>=16), col = L%16
// EXEC is all-ones here (no divergent control flow) as WMMA requires.
// ---------------------------------------------------------------------------
__device__ __forceinline__ void wg_gemm(float* __restrict__ D,
                                        const float* __restrict__ A,
                                        const float* __restrict__ B,
                                        int wave, int lane) {
  const int hi = lane >> 4;                  // lane half selects K sub-pair
  const int lm = lane & 15;

  // Hoist this wave's A band (16 rows x 128 K) into registers: 32 v2f frags.
  const float* arow = A + (wave * 16 + lm) * LDSS;
  v2f afrag[32];
#pragma unroll
  for (int kb = 0; kb < 32; ++kb)
    afrag[kb] = *(const v2f*)(arow + kb * 4 + 2 * hi);   // ds_load_b64

  for (int tn = 0; tn < 8; ++tn) {
    const float* bcol = B + tn * 16 + lm;
    v8f acc = {};
#pragma unroll
    for (int kb = 0; kb < 32; ++kb) {        // full unroll: afrag stays in regs
      const float* bp = bcol + (kb * 4 + 2 * hi) * LDSS;
      v2f bfrag;
      bfrag.x = bp[0];
      bfrag.y = bp[LDSS];
      acc = __builtin_amdgcn_wmma_f32_16x16x4_f32(
          /*neg_a=*/false, afrag[kb], /*neg_b=*/false, bfrag,
          /*c_mod=*/(short)0, acc, /*reuse_a=*/false, /*reuse_b=*/false);
    }
    float* drow = D + (wave * 16 + 8 * hi) * LDSS + tn * 16 + lm;
#pragma unroll
    for (int r = 0; r < 8; ++r) drow[r * LDSS] = acc[r];
  }
}

// ---------------------------------------------------------------------------
// One workgroup (256 threads / 8 waves) per batch matrix.
// Computes out[b] = exp(sqrt(A^2)) == U diag(exp(s)) U^T of the reference
// (SVD of symmetric A has s = |lambda|, U = Q).
// ---------------------------------------------------------------------------
__global__ __launch_bounds__(256)
void spd_untangent_kernel(const float* __restrict__ x,
                          float* __restrict__ out) {
  extern __shared__ float lds[];
  float* buf0 = lds;
  float* buf1 = lds + 1 * MATEL;
  float* buf2 = lds + 2 * MATEL;
  float* buf3 = lds + 3 * MATEL;
  float* red  = lds + 4 * MATEL;   // reduction scratch

  const int b    = blockIdx.x;
  const int tid  = threadIdx.x;
  const int wave = tid >> 5;
  const int lane = tid & 31;
  const float* xv = x + (size_t)b * VECLEN;

  // ---- 1. unvectorize: symmetric A into buf0 -----------------------------
  // triu (i<=j) flat index: i*128 - i*(i-1)/2 + (j-i)
  for (int e = tid; e < MATN * MATN; e += 256) {
    int i = e >> 7, j = e & 127;
    int r = i < j ? i : j;
    int c = i < j ? j : i;
    int idx = r * MATN - (r * (r - 1)) / 2 + (c - r);
    buf0[i * LDSS + j] = xv[idx];
  }
  if (tid == 0) red[0] = 0.0f;
  __syncthreads();

  // ---- 2. c = ||A||_F^2  (>= ||A^2||_2, Newton-Schulz normalizer) --------
  float part = 0.0f;
  for (int e = tid; e < MATN * MATN; e += 256) {
    float v = buf0[(e >> 7) * LDSS + (e & 127)];
    part += v * v;
  }
#pragma unroll
  for (int off = 16; off > 0; off >>= 1) part += __shfl_down(part, off, 32);
  if (lane == 0) atomicAdd(&red[0], part);   // ds_add_f32
  __syncthreads();
  const float c    = red[0] + 1e-30f;
  const float invc = 1.0f / c;
  __syncthreads();

  // ---- 3. B = A*A  -> buf1 ----------------------------------------------
  wg_gemm(buf1, buf0, buf0, wave, lane);
  __syncthreads();

  // ---- 4. Newton-Schulz: Y -> sqrt(B/c), Z -> inv-sqrt(B/c) --------------
  for (int e = tid; e < MATN * MATN; e += 256) {
    int i = e >> 7, j = e & 127;
    buf1[i * LDSS + j] *= invc;                       // Y0 = B/c
    buf0[i * LDSS + j] = (i == j) ? 1.0f : 0.0f;      // Z0 = I
  }
  __syncthreads();

  float* Y = buf1;
  float* Z = buf0;
  float* T = buf2;
  float* W = buf3;
  for (int it = 0; it < 14; ++it) {
    wg_gemm(T, Z, Y, wave, lane);                     // T = Z*Y
    __syncthreads();
    for (int e = tid; e < MATN * MATN; e += 256) {    // T <- (3I - T)/2
      int i = e >> 7, j = e & 127;
      float t = T[i * LDSS + j];
      T[i * LDSS + j] = ((i == j) ? 1.5f : 0.0f) - 0.5f * t;
    }
    __syncthreads();
    wg_gemm(W, Y, T, wave, lane);                     // Ynew = Y*T
    __syncthreads();
    wg_gemm(Y, T, Z, wave, lane);                     // Znew = T*Z (into old Y)
    __syncthreads();
    float* newY = W;
    float* newZ = Y;
    W = Z;                                            // rotate free buffers
    Y = newY;
    Z = newZ;
  }

  // ---- 5. X = sqrt(B)/2^6 = (sqrt(c)/64) * Y  (scaling step of expm) -----
  const float sc = sqrtf(c) * (1.0f / 64.0f);
  for (int e = tid; e < MATN * MATN; e += 256)
    Y[(e >> 7) * LDSS + (e & 127)] *= sc;
  __syncthreads();

  // ---- 6. exp(X) by 10-term Horner Taylor, then 6 squarings --------------
  float* X   = Y;
  float* E   = Z;
  float* Tmp = T;
  const int TERMS = 10;
  for (int e = tid; e < MATN * MATN; e += 256) {      // E = I + X/TERMS
    int i = e >> 7, j = e & 127;
    E[i * LDSS + j] =
        ((i == j) ? 1.0f : 0.0f) + X[i * LDSS + j] * (1.0f / (float)TERMS);
  }
  __syncthreads();
  for (int k = TERMS - 1; k >= 1; --k) {
    wg_gemm(Tmp, X, E, wave, lane);                   // Tmp = X*E
    __syncthreads();
    const float rk = 1.0f / (float)k;
    for (int e = tid; e < MATN * MATN; e += 256) {    // E = I + Tmp/k
      int i = e >> 7, j = e & 127;
      E[i * LDSS + j] = ((i == j) ? 1.0f : 0.0f) + Tmp[i * LDSS + j] * rk;
    }
    __syncthreads();
  }
  for (int s = 0; s < 6; ++s) {                       // E <- E^2, six times
    wg_gemm(Tmp, E, E, wave, lane);
    __syncthreads();
    float* t2 = E; E = Tmp; Tmp = t2;
  }

  // ---- 7. write result (f32, row-major 128x128) --------------------------
  float* dst = out + (size_t)b * (MATN * MATN);
  for (int e = tid; e < MATN * (MATN / 4); e += 256) {
    int i = e >> 5, jc = (e & 31) * 4;                // 16B-aligned in LDS
    float4 v = *(const float4*)(E + i * LDSS + jc);
    *(float4*)(dst + i * MATN + jc) = v;
  }
}

extern "C" void kernel_launch(void* const* d_in, const int* in_sizes, int n_in,
                              void* d_out, int out_size, void* d_ws, size_t ws_size,
                              hipStream_t stream) {
  (void)d_ws; (void)ws_size; (void)n_in; (void)out_size;
  const float* x = (const float*)d_in[0];
  float* out = (float*)d_out;
  const int batch = in_sizes[0] / VECLEN;             // 1024

  const size_t shmem = (size_t)(4 * MATEL + 32) * sizeof(float);  // ~264 KB
  // Opt in to >64KB dynamic LDS (WGP has 320KB on CDNA5).
  hipFuncSetAttribute((const void*)spd_untangent_kernel,
                      hipFuncAttributeMaxDynamicSharedMemorySize, (int)shmem);
  spd_untangent_kernel<<<batch, 256, shmem, stream>>>(x, out);
}